// MixedGeometricCoefficientAttention_34892314313427
// MI455X (gfx1250) — compile-verified
//
#include <hip/hip_runtime.h>
#include <math.h>

typedef float v2f __attribute__((ext_vector_type(2)));
typedef float v4f __attribute__((ext_vector_type(4)));
typedef float v8f __attribute__((ext_vector_type(8)));

#define BATCH   4
#define SEQ     2048
#define HEADS   8
#define MV      8
#define TOKENS  (BATCH*SEQ)
#define QK_SCALE 0.35355339059327376f   /* 8^-0.5 */
#define NQ      (BATCH*HEADS*SEQ*MV)    /* elements per [B,H,T,8] tensor */

// Per-wave LDS: 16x16 S/P tile at stride 18 (8B-aligned b64 reads,
// conflict-free) + 16-float row-stat broadcast buffer.
#define WTILE   (16*18)
#define WSTRIDE (WTILE + 16)

// ---------------------------------------------------------------------------
// Per-grade input projection: x[B,T,8] -> out[B,H,T,8]   (scale folded in)
// Grades: [0] d=1, [1..3] d=3, [4..6] d=3, [7] d=1.
// w1,w2 are (3, H*3) row-major; w0,w3 are (1,H).
// ---------------------------------------------------------------------------
__global__ __launch_bounds__(256)
void mgca_proj_kernel(const float* __restrict__ x,
                      const float* __restrict__ w0, const float* __restrict__ b0,
                      const float* __restrict__ w1, const float* __restrict__ b1,
                      const float* __restrict__ w2, const float* __restrict__ b2,
                      const float* __restrict__ w3, const float* __restrict__ b3,
                      float* __restrict__ out, float scale)
{
  int t = blockIdx.x * blockDim.x + threadIdx.x;
  if (t >= TOKENS) return;
  int b  = t / SEQ;
  int tt = t - b * SEQ;
  float xv[MV];
#pragma unroll
  for (int i = 0; i < MV; ++i) xv[i] = x[(size_t)t * MV + i];
#pragma unroll
  for (int h = 0; h < HEADS; ++h) {
    float o[MV];
    o[0] = xv[0] * w0[h] + b0[h];
#pragma unroll
    for (int j = 0; j < 3; ++j) {
      float s1 = b1[h*3 + j];
      float s2 = b2[h*3 + j];
#pragma unroll
      for (int i = 0; i < 3; ++i) {
        s1 += xv[1+i] * w1[i*(HEADS*3) + h*3 + j];
        s2 += xv[4+i] * w2[i*(HEADS*3) + h*3 + j];
      }
      o[1+j] = s1;
      o[4+j] = s2;
    }
    o[7] = xv[7] * w3[h] + b3[h];
    size_t base = (((size_t)(b*HEADS + h))*SEQ + tt) * MV;
#pragma unroll
    for (int i = 0; i < MV; ++i) out[base + i] = o[i] * scale;
  }
}

// ---------------------------------------------------------------------------
// Flash attention, one wave per 16-query tile.
//   S = Q*K^T : 2x V_WMMA_F32_16X16X4_F32  (head dim 8 = two K=4 chunks)
//   O += P*V  : 4x V_WMMA_F32_16X16X4_F32  (16 keys = four K=4 chunks)
// Softmax: S tile round-trips LDS once (needed anyway to transpose P into
// A-operand layout); after the transposed read each lane owns one row's
// 8 column values, so row max/sum are local trees + ONE shfl_xor(16).
// exp() is applied directly in A-operand registers. Per-row alpha (and
// final l) are re-broadcast into C/D row indexing via a 16-float LDS
// buffer (1 store + 2 b128 loads).
// ---------------------------------------------------------------------------
__global__ __launch_bounds__(256)
void mgca_flash_kernel(const float* __restrict__ Q,
                       const float* __restrict__ K,
                       const float* __restrict__ V,
                       const unsigned char* __restrict__ mask,
                       float* __restrict__ HO)
{
  __shared__ float smem[8 * WSTRIDE];
  const int lane = threadIdx.x & 31;
  const int wid  = threadIdx.x >> 5;
  const int tile = blockIdx.x * 8 + wid;       // 0..4095
  const int bh   = tile >> 7;                  // tile / (SEQ/16)
  const int q0   = (tile & 127) << 4;
  const int b    = bh >> 3;
  const int hg   = lane >> 4;                  // half-wave id (0/1)
  const int n    = lane & 15;
  const size_t base = (size_t)bh * SEQ * MV;

  float* pl   = smem + wid * WSTRIDE;          // 16x16 tile, stride 18
  float* abuf = pl + WTILE;                    // 16 row-stat slots

  // Q tile in A-operand layout: lane row m=n, chunk c covers mv {4c+2hg, 4c+1+2hg}
  v2f qa0, qa1;
  {
    const float* qp = Q + base + (size_t)(q0 + n) * MV;
    qa0 = *(const v2f*)(qp + 2*hg);
    qa1 = *(const v2f*)(qp + 4 + 2*hg);
  }

  v8f acc = {0.f,0.f,0.f,0.f,0.f,0.f,0.f,0.f};
  float m_loc = -INFINITY;                     // stats for row m = n (this lane)
  float l_loc = 0.f;

  const unsigned char* km = mask + (size_t)b * SEQ;

  for (int k0 = 0; k0 < SEQ; k0 += 16) {
    // K tile in B-operand layout: lane col n = key, VGPR pair = mv rows {2hg,2hg+1}
    const float* kp = K + base + (size_t)(k0 + n) * MV;
    v2f kb0 = *(const v2f*)(kp + 2*hg);
    v2f kb1 = *(const v2f*)(kp + 4 + 2*hg);

    // V tile in B-operand layout: rows = keys, cols = mv (cols 8..15 zero)
    v2f vb[4];
#pragma unroll
    for (int c = 0; c < 4; ++c) {
      const int kr = k0 + 4*c + 2*hg;
      float e0 = 0.f, e1 = 0.f;
      if (n < MV) {
        e0 = V[base + (size_t)kr * MV + n];
        e1 = V[base + (size_t)(kr + 1) * MV + n];
      }
      vb[c].x = e0;
      vb[c].y = e1;
    }

    v8f s = {0.f,0.f,0.f,0.f,0.f,0.f,0.f,0.f};
    s = __builtin_amdgcn_wmma_f32_16x16x4_f32(false, qa0, false, kb0, (short)0, s, false, false);
    s = __builtin_amdgcn_wmma_f32_16x16x4_f32(false, qa1, false, kb1, (short)0, s, false, false);

    // key padding mask: in C/D layout this lane's column is key k0+n
    if (km[k0 + n]) {
#pragma unroll
      for (int r = 0; r < 8; ++r) s[r] = -INFINITY;
    }

    // Store raw S row-major (row r+8hg, col n).
#pragma unroll
    for (int r = 0; r < 8; ++r) pl[(r + 8*hg) * 18 + n] = s[r];

    asm volatile("s_wait_dscnt 0x0" ::: "memory");   // cross-lane store->load

    // Transposed read: lane owns row m=n, columns {4c+2hg, 4c+1+2hg}.
    v2f sv[4];
#pragma unroll
    for (int c = 0; c < 4; ++c)
      sv[c] = *(const v2f*)(pl + n * 18 + 4*c + 2*hg);

    // Row max: local tree over 8 values + one half-wave exchange.
    float mx = fmaxf(fmaxf(fmaxf(sv[0].x, sv[0].y), fmaxf(sv[1].x, sv[1].y)),
                     fmaxf(fmaxf(sv[2].x, sv[2].y), fmaxf(sv[3].x, sv[3].y)));
    mx = fmaxf(mx, __shfl_xor(mx, 16, 32));

    float mnew  = fmaxf(m_loc, mx);
    float a     = (m_loc == mnew) ? 1.f : __expf(m_loc - mnew);
    m_loc       = mnew;
    abuf[n]     = a;                        // lanes n and n+16 write same value

    // P directly in A-operand registers.
    v2f pa[4];
#pragma unroll
    for (int c = 0; c < 4; ++c) {
      pa[c].x = __expf(sv[c].x - mnew);
      pa[c].y = __expf(sv[c].y - mnew);
    }

    // Row sum: local tree + one half-wave exchange.
    float ps = (pa[0].x + pa[0].y) + (pa[1].x + pa[1].y)
             + (pa[2].x + pa[2].y) + (pa[3].x + pa[3].y);
    ps += __shfl_xor(ps, 16, 32);
    l_loc = l_loc * a + ps;

    asm volatile("s_wait_dscnt 0x0" ::: "memory");   // alpha store->load

    // Rescale accumulator: acc[r] is row r+8hg -> alphas 8hg..8hg+7.
    v4f a0 = *(const v4f*)(abuf + 8*hg);
    v4f a1 = *(const v4f*)(abuf + 8*hg + 4);
#pragma unroll
    for (int r = 0; r < 4; ++r) { acc[r] *= a0[r]; acc[r+4] *= a1[r]; }

#pragma unroll
    for (int c = 0; c < 4; ++c)
      acc = __builtin_amdgcn_wmma_f32_16x16x4_f32(false, pa[c], false, vb[c], (short)0, acc, false, false);
  }

  // Broadcast final row sums into C/D row indexing and normalize.
  abuf[n] = l_loc;
  asm volatile("s_wait_dscnt 0x0" ::: "memory");
  v4f l0 = *(const v4f*)(abuf + 8*hg);
  v4f l1 = *(const v4f*)(abuf + 8*hg + 4);

  if (n < MV) {
#pragma unroll
    for (int r = 0; r < 4; ++r) {
      HO[base + (size_t)(q0 + r     + 8*hg) * MV + n] = acc[r]   / l0[r];
      HO[base + (size_t)(q0 + r + 4 + 8*hg) * MV + n] = acc[r+4] / l1[r];
    }
  }
}

// ---------------------------------------------------------------------------
// Output projection: head_out[B,H,T,8] -> out[B,T,8]
// w1,w2 are (H*3, 3) row-major; w0,w3 are (H,1).
// ---------------------------------------------------------------------------
__global__ __launch_bounds__(256)
void mgca_oproj_kernel(const float* __restrict__ HO,
                       const float* __restrict__ w0, const float* __restrict__ b0,
                       const float* __restrict__ w1, const float* __restrict__ b1,
                       const float* __restrict__ w2, const float* __restrict__ b2,
                       const float* __restrict__ w3, const float* __restrict__ b3,
                       float* __restrict__ out)
{
  int t = blockIdx.x * blockDim.x + threadIdx.x;
  if (t >= TOKENS) return;
  int b  = t / SEQ;
  int tt = t - b * SEQ;
  float ho[HEADS][MV];
#pragma unroll
  for (int h = 0; h < HEADS; ++h) {
    size_t base = (((size_t)(b*HEADS + h))*SEQ + tt) * MV;
#pragma unroll
    for (int i = 0; i < MV; ++i) ho[h][i] = HO[base + i];
  }
  float o[MV];
  float s0 = b0[0], s3 = b3[0];
#pragma unroll
  for (int h = 0; h < HEADS; ++h) {
    s0 += ho[h][0] * w0[h];
    s3 += ho[h][7] * w3[h];
  }
  o[0] = s0; o[7] = s3;
#pragma unroll
  for (int j = 0; j < 3; ++j) {
    float s1 = b1[j];
    float s2 = b2[j];
#pragma unroll
    for (int h = 0; h < HEADS; ++h) {
#pragma unroll
      for (int jp = 0; jp < 3; ++jp) {
        s1 += ho[h][1+jp] * w1[(h*3 + jp)*3 + j];
        s2 += ho[h][4+jp] * w2[(h*3 + jp)*3 + j];
      }
    }
    o[1+j] = s1;
    o[4+j] = s2;
  }
#pragma unroll
  for (int i = 0; i < MV; ++i) out[(size_t)t * MV + i] = o[i];
}

// ---------------------------------------------------------------------------
// d_in order: 0:x 1:mask 2-5:q_ws 6-9:q_bs 10-13:k_ws 14-17:k_bs
//             18-21:v_ws 22-25:v_bs 26-29:o_ws 30-33:o_bs
// Scratch: Q | K | V | HO  (4 x [B,H,T,8] fp32 = 8 MB)
// ---------------------------------------------------------------------------
extern "C" void kernel_launch(void* const* d_in, const int* in_sizes, int n_in,
                              void* d_out, int out_size, void* d_ws, size_t ws_size,
                              hipStream_t stream)
{
  const float* x = (const float*)d_in[0];
  const unsigned char* mask = (const unsigned char*)d_in[1];
  float* ws = (float*)d_ws;
  float* Q  = ws;
  float* Km = ws + (size_t)NQ;
  float* Vm = ws + (size_t)2 * NQ;
  float* HO = ws + (size_t)3 * NQ;

  dim3 blk(256), grd((TOKENS + 255) / 256);

  // Q projection with softmax scale folded in.
  mgca_proj_kernel<<<grd, blk, 0, stream>>>(x,
      (const float*)d_in[2], (const float*)d_in[6],
      (const float*)d_in[3], (const float*)d_in[7],
      (const float*)d_in[4], (const float*)d_in[8],
      (const float*)d_in[5], (const float*)d_in[9],
      Q, QK_SCALE);
  // K projection; k_mix coefficient ALPHA*rev*diag + (1-ALPHA) == 1.0 per component in Cl(3,0).
  mgca_proj_kernel<<<grd, blk, 0, stream>>>(x,
      (const float*)d_in[10], (const float*)d_in[14],
      (const float*)d_in[11], (const float*)d_in[15],
      (const float*)d_in[12], (const float*)d_in[16],
      (const float*)d_in[13], (const float*)d_in[17],
      Km, 1.0f);
  // V projection.
  mgca_proj_kernel<<<grd, blk, 0, stream>>>(x,
      (const float*)d_in[18], (const float*)d_in[22],
      (const float*)d_in[19], (const float*)d_in[23],
      (const float*)d_in[20], (const float*)d_in[24],
      (const float*)d_in[21], (const float*)d_in[25],
      Vm, 1.0f);

  // 4096 query tiles (16 queries each), 8 waves per block -> 512 blocks.
  mgca_flash_kernel<<<dim3(BATCH*HEADS*(SEQ/16)/8), blk, 0, stream>>>(Q, Km, Vm, mask, HO);

  mgca_oproj_kernel<<<grd, blk, 0, stream>>>(HO,
      (const float*)d_in[26], (const float*)d_in[30],
      (const float*)d_in[27], (const float*)d_in[31],
      (const float*)d_in[28], (const float*)d_in[32],
      (const float*)d_in[29], (const float*)d_in[33],
      (float*)d_out);
}